// RANSAC_37014028157311
// MI455X (gfx1250) — compile-verified
//
#include <hip/hip_runtime.h>

// ---------------------------------------------------------------------------
// RANSAC plane fit, MI455X (gfx1250, wave32).
//
//   K1 ransac_fit    : 3200 planes via cross-product (== null eigvec of the
//                      rank-3 Gram matrix the reference eigh's), replicating
//                      the reference's sign quirk (flip abc if b<0, keep d).
//                      Also zeroes counts + planesT padding (ws is poisoned).
//   K2 ransac_count  : inlier counting as GEMM  points[32768x4] x planes[4x112]
//                      using V_WMMA_F32_16X16X4_F32 (K=4 is an exact fit).
//                      Each wave keeps all 7 B-tiles in VGPRs, streams point
//                      tiles (one pass over the 12 MB L2-resident data array),
//                      abs/threshold/accumulate in VALU, one atomicAdd per
//                      plane column at wave end. M-loop unrolled x2 for MLP;
//                      WGP-scope prefetch of the next tile.
//   K3 ransac_select : sequential 100-step best-model scan, one wave32,
//                      lane==batch, shfl_xor max-reduce for global 'reason'.
// ---------------------------------------------------------------------------

typedef __attribute__((ext_vector_type(2))) float v2f;
typedef __attribute__((ext_vector_type(8))) float v8f;

#define RB          32          // batches
#define RN          32768       // points per batch
#define MAX_ITERS   100
#define NPAD        112         // 7 tiles of 16 (100 planes padded)
#define NTILES      7
#define THRESHOLD   0.05f
#define SIGNIF_LOGIT (-0.35667494393873245f)   // log(1 - 0.3)

#define WAVES_PER_BATCH 64
#define MTILES_PER_WAVE (RN / 16 / WAVES_PER_BATCH)   // 2048/64 = 32

// ---------------------------------------------------------------------------
// Kernel 1: fit 3200 planes; zero counts; zero planesT padding columns.
// planes  : [MAX_ITERS][RB][4]   (row layout for the select kernel)
// planesT : [RB][4][NPAD]        (K-major columns for WMMA B-tiles)
// counts  : [NPAD][RB] int
// ---------------------------------------------------------------------------
__global__ __launch_bounds__(256) void ransac_fit(
    const float* __restrict__ data, const int* __restrict__ idx,
    float* __restrict__ planes, float* __restrict__ planesT,
    int* __restrict__ counts) {
  const int tid = blockIdx.x * blockDim.x + threadIdx.x;

  if (tid < NPAD * RB) counts[tid] = 0;

  // zero planesT padding columns j in [100,112)
  const int PADW = NPAD - MAX_ITERS;                  // 12
  if (tid < RB * 4 * PADW) {
    int j = tid % PADW;
    int k = (tid / PADW) % 4;
    int b = tid / (PADW * 4);
    planesT[(b * 4 + k) * NPAD + MAX_ITERS + j] = 0.0f;
  }

  if (tid >= MAX_ITERS * RB) return;
  const int it = tid / RB;
  const int b  = tid % RB;

  const int* s = idx + (it * RB + b) * 3;
  float px[3], py[3], pz[3];
#pragma unroll
  for (int i = 0; i < 3; ++i) {
    const float* q = data + ((size_t)b * RN + (size_t)s[i]) * 3;
    px[i] = q[0]; py[i] = q[1]; pz[i] = q[2];
  }
  // plane through 3 points: n = (p1-p0) x (p2-p0), d = -n.p0
  const float e1x = px[1] - px[0], e1y = py[1] - py[0], e1z = pz[1] - pz[0];
  const float e2x = px[2] - px[0], e2y = py[2] - py[0], e2z = pz[2] - pz[0];
  float nx = e1y * e2z - e1z * e2y;
  float ny = e1z * e2x - e1x * e2z;
  float nz = e1x * e2y - e1y * e2x;
  float d  = -(nx * px[0] + ny * py[0] + nz * pz[0]);
  const float inv = rsqrtf(nx * nx + ny * ny + nz * nz + 1e-30f);
  nx *= inv; ny *= inv; nz *= inv; d *= inv;
  // reference quirk: sign flip applied to abc only, d untouched
  const float sgn = (ny < 0.0f) ? -1.0f : 1.0f;
  nx *= sgn; ny *= sgn; nz *= sgn;

  float* pr = planes + (size_t)(it * RB + b) * 4;
  pr[0] = nx; pr[1] = ny; pr[2] = nz; pr[3] = d;

  float* pt = planesT + (size_t)b * 4 * NPAD;
  pt[0 * NPAD + it] = nx;
  pt[1 * NPAD + it] = ny;
  pt[2 * NPAD + it] = nz;
  pt[3 * NPAD + it] = d;
}

// ---------------------------------------------------------------------------
// Kernel 2: inlier counting via V_WMMA_F32_16X16X4_F32.
//
// ISA f32 operand striping (CDNA5 ISA §7.12.2):
//   A (16x4, MxK): VGPR0 = K=0 (lanes 0-15) / K=2 (lanes 16-31)
//                  VGPR1 = K=1 (lanes 0-15) / K=3 (lanes 16-31); M = lane%16
//   B (4x16, KxN): VGPR0 = K=0 row (lanes 0-15) / K=2 row (lanes 16-31)
//                  VGPR1 = K=1 row / K=3 row; N = lane%16
//   D (16x16):     VGPR r, lanes 0-15: M=r, N=lane; lanes 16-31: M=r+8.
// => lane L's 8 D values are 8 point-rows of ONE plane column (L%16), so each
//    lane accumulates an integer inlier count for that plane; halves combined
//    with shfl_xor(16) before one atomicAdd per column.
// ---------------------------------------------------------------------------
__global__ __launch_bounds__(256) void ransac_count(
    const float* __restrict__ data, const float* __restrict__ planesT,
    int* __restrict__ counts) {
  const int wave   = (int)((blockIdx.x * blockDim.x + threadIdx.x) >> 5);
  const int lane   = threadIdx.x & 31;
  const int b      = wave / WAVES_PER_BATCH;
  const int wslot  = wave % WAVES_PER_BATCH;
  const int lane16 = lane & 15;
  const bool hi    = lane >= 16;

  // B-tiles for all 7 column groups, resident in VGPRs for the whole wave.
  const float* pt = planesT + (size_t)b * 4 * NPAD;
  const int k0 = hi ? 2 : 0;
  v2f Bt[NTILES];
#pragma unroll
  for (int t = 0; t < NTILES; ++t) {
    const int col = t * 16 + lane16;
    Bt[t].x = pt[(k0 + 0) * NPAD + col];
    Bt[t].y = pt[(k0 + 1) * NPAD + col];
  }

  int cnt[NTILES];
#pragma unroll
  for (int t = 0; t < NTILES; ++t) cnt[t] = 0;

  const float* dat = data + (size_t)b * RN * 3;
  const int m0 = wslot * MTILES_PER_WAVE * 16;

#pragma unroll 2
  for (int mt = 0; mt < MTILES_PER_WAVE; ++mt) {
    const int p = m0 + mt * 16 + lane16;
    const float* q = dat + (size_t)p * 3;
    const float x = q[0], y = q[1], z = q[2];
    if (mt + 1 < MTILES_PER_WAVE)
      __builtin_prefetch(q + 48, 0, 3);          // next tile, WGP scope
    v2f A;
    A.x = hi ? z : x;                             // K=2 : K=0
    A.y = hi ? 1.0f : y;                          // K=3 : K=1
#pragma unroll
    for (int t = 0; t < NTILES; ++t) {
      v8f c = {};
      v8f dv = __builtin_amdgcn_wmma_f32_16x16x4_f32(
          /*neg_a=*/false, A, /*neg_b=*/false, Bt[t],
          /*c_mod=*/(short)0, c, /*reuse_a=*/false, /*reuse_b=*/false);
      int s = 0;
#pragma unroll
      for (int r = 0; r < 8; ++r)
        s += (__builtin_fabsf(dv[r]) < THRESHOLD) ? 1 : 0;
      cnt[t] += s;
    }
  }

#pragma unroll
  for (int t = 0; t < NTILES; ++t) {
    const int c = cnt[t] + __shfl_xor(cnt[t], 16);
    const int iter = t * 16 + lane16;
    if (!hi && iter < MAX_ITERS) atomicAdd(&counts[iter * RB + b], c);
  }
}

// ---------------------------------------------------------------------------
// Kernel 3: sequential best-model scan. One wave32, lane == batch.
// ---------------------------------------------------------------------------
__global__ void ransac_select(const float* __restrict__ planes,
                              const int* __restrict__ counts,
                              float* __restrict__ out) {
  const int b = threadIdx.x;    // 0..31
  int best_ic = 0;
  float m0 = 0.f, m1 = 0.f, m2 = 0.f, m3 = 0.f;
  bool done = false;

  for (int i = 0; i < MAX_ITERS; ++i) {
    const int ic = counts[i * RB + b];
    const bool upd = (ic > best_ic) && !done;
    if (upd) {
      best_ic = ic;
      const float* pl = planes + (size_t)(i * RB + b) * 4;
      m0 = pl[0]; m1 = pl[1]; m2 = pl[2]; m3 = pl[3];
    }
    const float p    = (float)best_ic / (float)RN;
    const float prob = fminf(1.0f - p * p * p, 0.99f);
    float kk = SIGNIF_LOGIT / (logf(prob) + 1e-10f);
#pragma unroll
    for (int off = 16; off >= 1; off >>= 1)
      kk = fmaxf(kk, __shfl_xor(kk, off));       // global max over batches
    const int reason = min((int)kk, MAX_ITERS);
    done = done || (i + 1 >= reason);
  }
  out[b * 4 + 0] = m0;
  out[b * 4 + 1] = m1;
  out[b * 4 + 2] = m2;
  out[b * 4 + 3] = m3;
}

// ---------------------------------------------------------------------------
extern "C" void kernel_launch(void* const* d_in, const int* in_sizes, int n_in,
                              void* d_out, int out_size, void* d_ws, size_t ws_size,
                              hipStream_t stream) {
  const float* data = (const float*)d_in[0];   // [32][32768][3] f32
  const int*   idx  = (const int*)d_in[1];     // [100][32][3] i32

  float* planes  = (float*)d_ws;                       // 12800 f32
  float* planesT = planes + MAX_ITERS * RB * 4;        // 14336 f32
  int*   counts  = (int*)(planesT + RB * 4 * NPAD);    // 3584 i32

  // K1: 3584 work items (max of plane-count / counts-zero / pad-zero needs)
  ransac_fit<<<(NPAD * RB + 255) / 256, 256, 0, stream>>>(
      data, idx, planes, planesT, counts);

  // K2: 32 batches x 64 waves x 32 lanes = 65536 threads -> 256 blocks
  ransac_count<<<(RB * WAVES_PER_BATCH * 32) / 256, 256, 0, stream>>>(
      data, planesT, counts);

  // K3: single wave
  ransac_select<<<1, 32, 0, stream>>>(planes, counts, (float*)d_out);
}